// HeatMap_19542101197245
// MI455X (gfx1250) — compile-verified
//
#include <hip/hip_runtime.h>

// HeatMap scatter-max:
//   out[b, 0, :, :] = 0
//   for each landmark (b,n): clamp, split int/frac, write 13x13 patch of
//   1/sqrt(1 + ||off - frac||^2 + 1e-6) with max-combine.
//
// MI455X reasoning: output is 256 MB f32 -> store-bandwidth bound
// (~11 us at 23.3 TB/s). Compute (4.2M rsqrt) is negligible. No GEMM
// structure -> WMMA not applicable; the wins are b128 zero-fill stores and
// non-returning global_atomic_max_i32 (exact for positive floats vs zero
// background).

#define IMG   512
#define HALF  6
#define PATCH 13
#define P2    (PATCH * PATCH)   // 169
#define NB    256
#define NL    98

__global__ __launch_bounds__(256) void heatmap_zero_kernel(float4* __restrict__ out,
                                                           int n4) {
    int i      = blockIdx.x * blockDim.x + threadIdx.x;
    int stride = gridDim.x * blockDim.x;
    const float4 z = make_float4(0.0f, 0.0f, 0.0f, 0.0f);
    // b128 stores, grid-stride to cover 16.7M float4s
    for (; i < n4; i += stride) out[i] = z;
}

__global__ __launch_bounds__(256) void heatmap_scatter_kernel(
    const float* __restrict__ lmk,   // [B, N, 2] (row, col)
    float* __restrict__ out,         // [B, 512, 512]
    int total)                       // B*N*P2
{
    int idx = blockIdx.x * blockDim.x + threadIdx.x;
    if (idx >= total) return;

    int p  = idx % P2;     // patch element (row-major over (dy,dx))
    int ln = idx / P2;     // flat landmark index  b*NL + n
    int b  = ln / NL;

    // All 169 threads of a landmark hit the same 8 input bytes -> L0/L2 hit.
    float row = lmk[ln * 2 + 0];
    float col = lmk[ln * 2 + 1];

    const float lo = (float)HALF;              // 6
    const float hi = (float)(IMG - 1 - HALF);  // 505
    row = fminf(fmaxf(row, lo), hi);
    col = fminf(fmaxf(col, lo), hi);

    int   iy = (int)row;                // trunc == floor (row >= 6)
    int   ix = (int)col;
    float dy = row - (float)iy;         // subpixel residual
    float dx = col - (float)ix;

    int oy = p / PATCH - HALF;          // -6..6
    int ox = p % PATCH - HALF;

    float sy = (float)oy - dy;
    float sx = (float)ox - dx;
    float s  = fmaf(sy, sy, fmaf(sx, sx, 1.0f + 1e-6f));
    float val = 1.0f / sqrtf(s);        // in (0, 1]

    int y = iy + oy;                    // guaranteed 0..511 by clamp
    int x = ix + ox;

    // Positive floats order identically as signed ints; background is 0.
    // -> exact scatter-max via non-returning global_atomic_max_i32.
    int* addr = (int*)(out + ((size_t)b * IMG + (size_t)y) * IMG + (size_t)x);
    atomicMax(addr, __float_as_int(val));
}

extern "C" void kernel_launch(void* const* d_in, const int* in_sizes, int n_in,
                              void* d_out, int out_size, void* d_ws, size_t ws_size,
                              hipStream_t stream) {
    (void)in_sizes; (void)n_in; (void)d_ws; (void)ws_size;

    const float* lmk = (const float*)d_in[0];   // [256, 98, 2] f32
    float*       out = (float*)d_out;           // [256, 1, 512, 512] f32

    // --- Pass 1: zero the 256 MB output with b128 stores ---
    int n4 = out_size / 4;                      // 16,777,216 float4s
    int zthreads = 256;
    int zblocks  = 4096;                        // grid-stride; saturates all WGPs
    heatmap_zero_kernel<<<zblocks, zthreads, 0, stream>>>((float4*)out, n4);

    // --- Pass 2: scatter-max the 13x13 patches ---
    int total  = NB * NL * P2;                  // 4,239,872
    int blocks = (total + 255) / 256;
    heatmap_scatter_kernel<<<blocks, 256, 0, stream>>>(lmk, out, total);
}